// DiagonalLayer_3410204033731
// MI455X (gfx1250) — compile-verified
//
#include <hip/hip_runtime.h>
#include <hip/hip_bf16.h>

// out[b,g] = sum_{f<3} x[b, g*3+f] * w[g*3+f] + bias[g]
// HBM-streaming kernel: ~605 MB moved for 0.227 GFLOP (AI ~0.38) ->
// bandwidth floor ~26 us at 23.3 TB/s. fp32 block-diagonal contraction:
// WMMA inapplicable without 16x FLOP inflation. Optimize the stream:
// NT cache hints, branch-free hot path, immediate-offset addressing.

#define NUM_GENES 9229
#define NUM_FEATURES 3
#define BATCH 4096

constexpr int GPT  = 4;            // genes per thread (register-cached weights)
constexpr int TPB  = 256;          // 8 waves (wave32)
constexpr int GPB  = GPT * TPB;    // 1024 genes per block
constexpr int ROWS = 8;            // batch rows per block (weight reuse 8x)

constexpr int MAIN_GENES = (NUM_GENES / GPB) * GPB;   // 9216 = 9 blocks, exact
constexpr int TAIL_GENES = NUM_GENES - MAIN_GENES;    // 13

constexpr int XSTRIDE = NUM_GENES * NUM_FEATURES;     // 27687 floats per x row
constexpr int OSTRIDE = NUM_GENES;                    // 9229 floats per out row

// ---------------- Main kernel: genes [0, 9216), fully branch-free ----------
__global__ __launch_bounds__(TPB)
void diag_gene_main(const float* __restrict__ x,
                    const float* __restrict__ w,
                    const float* __restrict__ bias,
                    float* __restrict__ out)
{
    const int g0   = blockIdx.x * GPB + (int)threadIdx.x * GPT;  // always < 9216
    const int row0 = blockIdx.y * ROWS;

    // Register-cache 12 weights + 4 biases (L2-hot, reused ROWS times).
    float wreg[GPT][NUM_FEATURES];
    float breg[GPT];
#pragma unroll
    for (int i = 0; i < GPT; ++i) {
#pragma unroll
        for (int f = 0; f < NUM_FEATURES; ++f)
            wreg[i][f] = w[(g0 + i) * NUM_FEATURES + f];
        breg[i] = bias[g0 + i];
    }

    // One base pointer each; all row/gene offsets are compile-time constants,
    // so every access below is a single imm-offset vmem op off one VGPR pair.
    const float* __restrict__ xbase =
        x + (size_t)row0 * XSTRIDE + (size_t)g0 * NUM_FEATURES;
    float* __restrict__ obase = out + (size_t)row0 * OSTRIDE + g0;

#pragma unroll
    for (int r = 0; r < ROWS; ++r) {
        float acc[GPT];
#pragma unroll
        for (int i = 0; i < GPT; ++i) acc[i] = breg[i];

        // x stream is read exactly once -> non-temporal (don't pollute L2,
        // keep it for the hot 144 KB of weights/bias).
#pragma unroll
        for (int i = 0; i < GPT; ++i) {
#pragma unroll
            for (int f = 0; f < NUM_FEATURES; ++f) {
                const float xv = __builtin_nontemporal_load(
                    &xbase[r * XSTRIDE + i * NUM_FEATURES + f]);
                acc[i] = fmaf(xv, wreg[i][f], acc[i]);
            }
        }

#pragma unroll
        for (int i = 0; i < GPT; ++i)
            __builtin_nontemporal_store(acc[i], &obase[r * OSTRIDE + i]);
    }
}

// ---------------- Tail kernel: genes [9216, 9229) x all rows ---------------
__global__ __launch_bounds__(TPB)
void diag_gene_tail(const float* __restrict__ x,
                    const float* __restrict__ w,
                    const float* __restrict__ bias,
                    float* __restrict__ out)
{
    const int idx = blockIdx.x * TPB + (int)threadIdx.x;
    if (idx >= BATCH * TAIL_GENES) return;

    const int row = idx / TAIL_GENES;
    const int g   = MAIN_GENES + (idx - row * TAIL_GENES);

    const float* __restrict__ xr =
        x + (size_t)row * XSTRIDE + (size_t)g * NUM_FEATURES;

    float acc = bias[g];
#pragma unroll
    for (int f = 0; f < NUM_FEATURES; ++f) {
        const float xv = __builtin_nontemporal_load(&xr[f]);
        acc = fmaf(xv, w[g * NUM_FEATURES + f], acc);
    }
    __builtin_nontemporal_store(acc, &out[(size_t)row * OSTRIDE + g]);
}

extern "C" void kernel_launch(void* const* d_in, const int* in_sizes, int n_in,
                              void* d_out, int out_size, void* d_ws, size_t ws_size,
                              hipStream_t stream) {
    (void)in_sizes; (void)n_in; (void)d_ws; (void)ws_size; (void)out_size;

    const float* x    = (const float*)d_in[0];  // [BATCH, NUM_GENES*NUM_FEATURES]
    const float* wgt  = (const float*)d_in[1];  // [NUM_GENES*NUM_FEATURES]
    const float* bias = (const float*)d_in[2];  // [NUM_GENES]
    float* out        = (float*)d_out;          // [BATCH, NUM_GENES]

    // Main: 9 gene-blocks x 512 row-groups = 4608 branch-free blocks.
    dim3 grid_main(MAIN_GENES / GPB, BATCH / ROWS);
    diag_gene_main<<<grid_main, TPB, 0, stream>>>(x, wgt, bias, out);

    // Tail: 13 genes x 4096 rows = 53248 threads -> 208 blocks.
    const int tail_threads = BATCH * TAIL_GENES;
    dim3 grid_tail((tail_threads + TPB - 1) / TPB);
    diag_gene_tail<<<grid_tail, TPB, 0, stream>>>(x, wgt, bias, out);
}